// GAT_63161789055512
// MI455X (gfx1250) — compile-verified
//
#include <hip/hip_runtime.h>

// ---------------------------------------------------------------------------
// GAT (3-layer) for MI455X / gfx1250. GEMMs via V_WMMA_F32_16X16X4_F32
// (exact f32, wave32). Each wave computes a 16x64 output strip: one A-frag
// load feeds 4 WMMAs (4 independent accumulator chains -> latency hiding,
// 1.25 loads/WMMA). Edge softmax via ordered-uint atomicMax + f32 atomics;
// all per-node arrays are L2-resident (192 MB L2 >> 20 MB feature maps).
// ---------------------------------------------------------------------------

typedef __attribute__((ext_vector_type(2))) float v2f;
typedef __attribute__((ext_vector_type(8))) float v8f;

// ---- ordered-float <-> uint monotone encoding for atomic max ---------------
__device__ __forceinline__ unsigned f2ord(float f) {
    unsigned u = __float_as_uint(f);
    return (u & 0x80000000u) ? ~u : (u | 0x80000000u);
}
__device__ __forceinline__ float ord2f(unsigned u) {
    return (u & 0x80000000u) ? __uint_as_float(u & 0x7FFFFFFFu)
                             : __uint_as_float(~u);
}

// ---------------------------------------------------------------------------
// C[n, m] = sum_k A[n, k] * W[m, k]   (NT GEMM, f32, N%16==0, M%64==0,
// K%4==0).  One wave32 computes a 16x64 strip of C: 4 WMMA tiles sharing one
// A fragment per k-step.
//
// Fragment layouts (ISA 7.12.2):
//   A 16x4 f32 : lanes 0-15 row M=lane, VGPR0=K0,VGPR1=K1;
//                lanes 16-31 same rows, VGPR0=K2,VGPR1=K3.
//   B 4x16 f32 : VGPR v = row K=v (lanes 0-15) / K=v+2 (lanes 16-31),
//                column = lane&15.  (Same addressing as A for NT.)
//   C/D 16x16  : VGPR r = row r (lanes 0-15) / r+8 (lanes 16-31),
//                column = lane&15.
// ---------------------------------------------------------------------------
__global__ __launch_bounds__(256) void gemm_nt_wmma(
    const float* __restrict__ A, const float* __restrict__ W,
    float* __restrict__ C, int K, int M, int tiles_m64, int tiles_total)
{
    int wave = blockIdx.x * (blockDim.x >> 5) + (threadIdx.x >> 5);
    if (wave >= tiles_total) return;               // wave-uniform branch
    int tm = wave % tiles_m64;                     // 64-wide column strip
    int tn = wave / tiles_m64;

    int lane = threadIdx.x & 31;
    int half = lane >> 4;                          // 0: K0/K1, 1: K2/K3
    int l15  = lane & 15;

    const float* ap = A + (tn * 16 + l15) * (size_t)K + half * 2;
    const float* bp = W + (tm * 64 + l15) * (size_t)K + half * 2;
    const size_t bstep = (size_t)16 * K;           // next 16-col tile of W

    v8f acc0 = {}, acc1 = {}, acc2 = {}, acc3 = {};
    #pragma unroll 2
    for (int k = 0; k < K; k += 4) {
        v2f a  = *(const v2f*)(ap + k);
        v2f b0 = *(const v2f*)(bp + k);
        v2f b1 = *(const v2f*)(bp + bstep + k);
        v2f b2 = *(const v2f*)(bp + 2 * bstep + k);
        v2f b3 = *(const v2f*)(bp + 3 * bstep + k);
        acc0 = __builtin_amdgcn_wmma_f32_16x16x4_f32(false, a, false, b0,
                                                     (short)0, acc0, false, false);
        acc1 = __builtin_amdgcn_wmma_f32_16x16x4_f32(false, a, false, b1,
                                                     (short)0, acc1, false, false);
        acc2 = __builtin_amdgcn_wmma_f32_16x16x4_f32(false, a, false, b2,
                                                     (short)0, acc2, false, false);
        acc3 = __builtin_amdgcn_wmma_f32_16x16x4_f32(false, a, false, b3,
                                                     (short)0, acc3, false, false);
    }

    float* cp = C + (size_t)(tn * 16 + half * 8) * M + tm * 64 + l15;
    #pragma unroll
    for (int r = 0; r < 8; ++r) {
        cp[(size_t)r * M +  0] = acc0[r];
        cp[(size_t)r * M + 16] = acc1[r];
        cp[(size_t)r * M + 32] = acc2[r];
        cp[(size_t)r * M + 48] = acc3[r];
    }
}

// alpha_s[n,h] = sum_c h[n,h,c]*a_src[h,c] ; alpha_d likewise
__global__ __launch_bounds__(256) void alpha_kernel(
    const float* __restrict__ h, const float* __restrict__ a_src,
    const float* __restrict__ a_dst, float* __restrict__ as_,
    float* __restrict__ ad_, int N, int H, int C)
{
    int idx = blockIdx.x * blockDim.x + threadIdx.x;    // n*H + hd
    if (idx >= N * H) return;
    int hd = idx % H;
    const float* hp = h + (size_t)(idx / H) * H * C + hd * C;
    float s = 0.f, d = 0.f;
    for (int c = 0; c < C; ++c) {
        float v = hp[c];
        s += v * a_src[hd * C + c];
        d += v * a_dst[hd * C + c];
    }
    as_[idx] = s;
    ad_[idx] = d;
}

// pass 1: e = leaky_relu(as[src]+ad[dst]); store e; segment max into mkey[dst]
__global__ __launch_bounds__(256) void edge_lrelu_max(
    const int* __restrict__ src, const int* __restrict__ dst, int E, int N,
    int H, const float* __restrict__ as_, const float* __restrict__ ad_,
    float* __restrict__ e_buf, unsigned* __restrict__ mkey)
{
    int idx = blockIdx.x * blockDim.x + threadIdx.x;
    if (idx >= (E + N) * H) return;
    int e = idx / H, hd = idx - e * H;
    int s = (e < E) ? src[e] : (e - E);
    int d = (e < E) ? dst[e] : (e - E);
    float v = as_[s * H + hd] + ad_[d * H + hd];
    v = (v > 0.f) ? v : 0.2f * v;                  // leaky_relu, slope 0.2
    e_buf[idx] = v;
    atomicMax(&mkey[d * H + hd], f2ord(v));
}

// pass 2: p = exp(e - m[dst]); store p; segment sum into denom[dst]
__global__ __launch_bounds__(256) void edge_expsum(
    const int* __restrict__ dst, int E, int N, int H,
    const unsigned* __restrict__ mkey, float* __restrict__ e_buf,
    float* __restrict__ denom)
{
    int idx = blockIdx.x * blockDim.x + threadIdx.x;
    if (idx >= (E + N) * H) return;
    int e = idx / H, hd = idx - e * H;
    int d = (e < E) ? dst[e] : (e - E);
    float p = __expf(e_buf[idx] - ord2f(mkey[d * H + hd]));
    e_buf[idx] = p;
    atomicAdd(&denom[d * H + hd], p);
}

// pass 3: alpha = p / (denom[dst] + 1e-16)
__global__ __launch_bounds__(256) void edge_norm(
    const int* __restrict__ dst, int E, int N, int H,
    const float* __restrict__ denom, float* __restrict__ e_buf)
{
    int idx = blockIdx.x * blockDim.x + threadIdx.x;
    if (idx >= (E + N) * H) return;
    int e = idx / H, hd = idx - e * H;
    int d = (e < E) ? dst[e] : (e - E);
    e_buf[idx] = e_buf[idx] / (denom[d * H + hd] + 1e-16f);
}

// pass 4: acc[dst, h, c] += h[src, h, c] * alpha[e, h]  (float4 per thread)
__global__ __launch_bounds__(256) void edge_agg4(
    const int* __restrict__ src, const int* __restrict__ dst, int E, int N,
    int HC, int C, int H, const float* __restrict__ h,
    const float* __restrict__ alpha, float* __restrict__ acc)
{
    int q = HC >> 2;                               // float4 slots per edge
    int idx = blockIdx.x * blockDim.x + threadIdx.x;   // e*q + t
    if (idx >= (E + N) * q) return;
    int e = idx / q, t4 = (idx - e * q) << 2;
    int s = (e < E) ? src[e] : (e - E);
    int d = (e < E) ? dst[e] : (e - E);
    float a = alpha[e * H + (t4 / C)];             // 4 channels share one head
    const float4 hv = *(const float4*)(h + (size_t)s * HC + t4);
    float* ap = acc + (size_t)d * HC + t4;
    atomicAdd(ap + 0, hv.x * a);
    atomicAdd(ap + 1, hv.y * a);
    atomicAdd(ap + 2, hv.z * a);
    atomicAdd(ap + 3, hv.w * a);
}

// out = acc + bias [; relu]
__global__ __launch_bounds__(256) void finalize_kernel(
    const float* __restrict__ acc, const float* __restrict__ bias,
    float* __restrict__ out, int N, int HC, int do_relu)
{
    int idx = blockIdx.x * blockDim.x + threadIdx.x;
    if (idx >= N * HC) return;
    float v = acc[idx] + bias[idx % HC];
    if (do_relu) v = fmaxf(v, 0.f);
    out[idx] = v;
}

// ---------------------------------------------------------------------------
static void run_gat_layer(const float* Xin, int K, const float* W,
                          const float* a_src, const float* a_dst,
                          const float* bias, int H, int C, int do_relu,
                          float* Xout, const int* esrc, const int* edst,
                          int N, int E, float* h_buf, float* as_buf,
                          float* ad_buf, unsigned* mkey, float* denom,
                          float* e_buf, float* acc, hipStream_t stream)
{
    const int M = H * C;
    const int Etot = E + N;

    int tiles = (N / 16) * (M / 64);
    gemm_nt_wmma<<<(tiles + 7) / 8, 256, 0, stream>>>(Xin, W, h_buf, K, M,
                                                      M / 64, tiles);
    int nh = N * H;
    alpha_kernel<<<(nh + 255) / 256, 256, 0, stream>>>(h_buf, a_src, a_dst,
                                                       as_buf, ad_buf, N, H, C);
    hipMemsetAsync(mkey, 0, (size_t)nh * sizeof(unsigned), stream);
    hipMemsetAsync(denom, 0, (size_t)nh * sizeof(float), stream);
    hipMemsetAsync(acc, 0, (size_t)N * M * sizeof(float), stream);

    int eh = Etot * H;
    edge_lrelu_max<<<(eh + 255) / 256, 256, 0, stream>>>(esrc, edst, E, N, H,
                                                         as_buf, ad_buf, e_buf,
                                                         mkey);
    edge_expsum<<<(eh + 255) / 256, 256, 0, stream>>>(edst, E, N, H, mkey,
                                                      e_buf, denom);
    edge_norm<<<(eh + 255) / 256, 256, 0, stream>>>(edst, E, N, H, denom,
                                                    e_buf);
    long long ehq = (long long)Etot * (M >> 2);
    edge_agg4<<<(unsigned)((ehq + 255) / 256), 256, 0, stream>>>(
        esrc, edst, E, N, M, C, H, h_buf, e_buf, acc);

    int nhc = N * M;
    finalize_kernel<<<(nhc + 255) / 256, 256, 0, stream>>>(acc, bias, Xout, N,
                                                           M, do_relu);
}

extern "C" void kernel_launch(void* const* d_in, const int* in_sizes, int n_in,
                              void* d_out, int out_size, void* d_ws,
                              size_t ws_size, hipStream_t stream)
{
    const float* x  = (const float*)d_in[0];
    const int*   ei = (const int*)d_in[1];
    const float* W0 = (const float*)d_in[2];
    const float* as0 = (const float*)d_in[3];
    const float* ad0 = (const float*)d_in[4];
    const float* b0 = (const float*)d_in[5];
    const float* W1 = (const float*)d_in[6];
    const float* as1 = (const float*)d_in[7];
    const float* ad1 = (const float*)d_in[8];
    const float* b1 = (const float*)d_in[9];
    const float* W2 = (const float*)d_in[10];
    const float* as2 = (const float*)d_in[11];
    const float* ad2 = (const float*)d_in[12];
    const float* b2 = (const float*)d_in[13];

    const int F_in = 512, H = 8, C = 32, HC = H * C, OUT = 128;
    const int N = in_sizes[0] / F_in;       // 20000 (divisible by 16)
    const int E = in_sizes[1] / 2;          // 320000
    const int* esrc = ei;
    const int* edst = ei + E;

    // workspace layout (floats)
    float* ws = (float*)d_ws;
    float* h_buf  = ws;                      // N*256
    float* acc    = h_buf + (size_t)N * HC;  // N*256
    float* X1     = acc + (size_t)N * HC;    // N*256 (layer0 output)
    float* as_buf = X1 + (size_t)N * HC;     // N*8
    float* ad_buf = as_buf + (size_t)N * H;  // N*8
    unsigned* mkey = (unsigned*)(ad_buf + (size_t)N * H);  // N*8
    float* denom  = (float*)(mkey + (size_t)N * H);        // N*8
    float* e_buf  = denom + (size_t)N * H;   // (E+N)*8

    float* out   = (float*)d_out;            // [0, N*256): h_pen
    float* out2  = out + (size_t)N * HC;     // [N*256, +N*128): final

    // layer 0: F_in -> H*C, relu
    run_gat_layer(x, F_in, W0, as0, ad0, b0, H, C, /*relu=*/1, X1, esrc, edst,
                  N, E, h_buf, as_buf, ad_buf, mkey, denom, e_buf, acc, stream);
    // layer 1: H*C -> H*C, relu; result is h_pen, written straight to d_out
    run_gat_layer(X1, HC, W1, as1, ad1, b1, H, C, /*relu=*/1, out, esrc, edst,
                  N, E, h_buf, as_buf, ad_buf, mkey, denom, e_buf, acc, stream);
    // layer 2: H*C -> OUT, 1 head, no concat/relu (mean over 1 head = identity)
    run_gat_layer(out, HC, W2, as2, ad2, b2, /*H=*/1, /*C=*/OUT, /*relu=*/0,
                  out2, esrc, edst, N, E, h_buf, as_buf, ad_buf, mkey, denom,
                  e_buf, acc, stream);
}